// CenterLoss_9740985827850
// MI455X (gfx1250) — compile-verified
//
#include <hip/hip_runtime.h>

// CDNA5 / gfx1250: f32 WMMA 16x16x4, wave32.
typedef __attribute__((ext_vector_type(2))) float v2f;
typedef __attribute__((ext_vector_type(8))) float v8f;

#define BATCH 16
#define NPRED 4096
#define MGT   512

// Zero the two scalar accumulators each launch (harness poisons ws once; graph
// replays require deterministic re-init).
__global__ void cl_init(float* acc) {
    if (threadIdx.x < 8) acc[threadIdx.x] = 0.0f;
}

// For each row r of Arows (per batch), compute min_c ||A_r - B_c||^2 over ALL
// C columns, multiply by wRow[r], and atomically accumulate the sum.
// One wave handles a 16-row tile; dot products via V_WMMA_F32_16X16X4_F32
// with K=3 padded to 4:  dist = |a|^2 + |b|^2 - 2 a.b  (exact f32 algebra path).
template<typename WT>
__global__ __launch_bounds__(256) void cl_chamfer(
    const float* __restrict__ Arows,   // BATCH x R x 3
    const float* __restrict__ Bcols,   // BATCH x C x 3
    const WT*    __restrict__ wRow,    // BATCH x R
    float*       __restrict__ accum,   // single f32 accumulator
    int R, int C)
{
    __shared__ float pn2_lds[128];               // 8 waves * 16 row norms
    const int tid  = threadIdx.x;
    const int wave = tid >> 5;
    const int lane = tid & 31;
    const int lo   = lane & 15;                  // column / row-in-tile index
    const int hi   = lane >> 4;                  // K-half selector

    const int tilesPerBatch = R >> 4;
    const int tileId = blockIdx.x * 8 + wave;
    const int b  = tileId / tilesPerBatch;
    const int r0 = (tileId % tilesPerBatch) << 4;

    // ---- A operand: row r0+lo. Lanes 0-15 carry K=0,1; lanes 16-31 carry K=2,pad0.
    const float* ar = Arows + ((size_t)b * R + (r0 + lo)) * 3;
    const float p0 = ar[0], p1 = ar[1], p2 = ar[2];
    v2f amat;
    amat.x = hi ? p2   : p0;
    amat.y = hi ? 0.0f : p1;
    const float pn2 = p0 * p0 + p1 * p1 + p2 * p2;
    if (hi == 0) pn2_lds[wave * 16 + lo] = pn2;
    __syncthreads();

    // Each lane post-processes D rows (8*hi + j), j=0..7 -> preload those norms.
    float pn2r[8];
#pragma unroll
    for (int j = 0; j < 8; ++j) pn2r[j] = pn2_lds[wave * 16 + 8 * hi + j];

    float dmin[8];
#pragma unroll
    for (int j = 0; j < 8; ++j) dmin[j] = 3.402823e38f;

    for (int c0 = 0; c0 < C; c0 += 16) {
        // ---- B operand: column c0+lo, same half-split K layout as A.
        const float* br = Bcols + ((size_t)b * C + (c0 + lo)) * 3;
        const float g0 = br[0], g1 = br[1], g2 = br[2];
        v2f bmat;
        bmat.x = hi ? g2   : g0;
        bmat.y = hi ? 0.0f : g1;
        const float gn2 = g0 * g0 + g1 * g1 + g2 * g2;

        v8f cz = {};
        // D = A(16x4) x B(4x16) + 0 : exact f32 dot products.
        v8f d = __builtin_amdgcn_wmma_f32_16x16x4_f32(
            /*neg_a=*/false, amat, /*neg_b=*/false, bmat,
            /*c_mod=*/(short)0, cz, /*reuse_a=*/false, /*reuse_b=*/false);

#pragma unroll
        for (int j = 0; j < 8; ++j) {
            // D layout: VGPR j, this lane -> row (r0 + 8*hi + j), col (c0 + lo).
            float dist = pn2r[j] + gn2 - 2.0f * d[j];
            dist = fmaxf(dist, 0.0f);            // guard tiny negative round-off
            dmin[j] = fminf(dmin[j], dist);
        }
    }

    // Min across the 16 column-lanes inside each half (xor 1,2,4,8 stays in-half).
#pragma unroll
    for (int j = 0; j < 8; ++j) {
        float v = dmin[j];
        v = fminf(v, __shfl_xor(v, 1, 32));
        v = fminf(v, __shfl_xor(v, 2, 32));
        v = fminf(v, __shfl_xor(v, 4, 32));
        v = fminf(v, __shfl_xor(v, 8, 32));
        dmin[j] = v;
    }

    // One lane per half accumulates its 8 rows' weighted mins.
    if (lo == 0) {
        float s = 0.0f;
#pragma unroll
        for (int j = 0; j < 8; ++j) {
            const int r = r0 + 8 * hi + j;
            s += dmin[j] * (float)wRow[(size_t)b * R + r];
        }
        atomicAdd(accum, s);
    }
}

__global__ __launch_bounds__(256) void cl_final(
    const float* __restrict__ accum,   // [0]=sum(dist1*obj), [1]=sum(dist2*mask)
    const int*   __restrict__ obj,
    const float* __restrict__ mask,
    float*       __restrict__ out)
{
    __shared__ float r0s[256], r1s[256];
    const int tid = threadIdx.x;
    float sobj = 0.0f;
    for (int i = tid; i < BATCH * NPRED; i += 256) sobj += (float)obj[i];
    float smask = 0.0f;
    for (int i = tid; i < BATCH * MGT; i += 256) smask += mask[i];
    r0s[tid] = sobj; r1s[tid] = smask;
    __syncthreads();
    for (int s = 128; s > 0; s >>= 1) {
        if (tid < s) { r0s[tid] += r0s[tid + s]; r1s[tid] += r1s[tid + s]; }
        __syncthreads();
    }
    if (tid == 0) {
        out[0] = accum[0] / (r0s[0] + 1e-6f) + accum[1] / (r1s[0] + 1e-6f);
    }
}

extern "C" void kernel_launch(void* const* d_in, const int* in_sizes, int n_in,
                              void* d_out, int out_size, void* d_ws, size_t ws_size,
                              hipStream_t stream) {
    const float* pred = (const float*)d_in[0];   // (16, 4096, 3) f32
    const float* gt   = (const float*)d_in[1];   // (16, 512, 3)  f32
    const float* mask = (const float*)d_in[2];   // (16, 512)     f32
    const int*   obj  = (const int*)d_in[3];     // (16, 4096)    i32
    float* out = (float*)d_out;
    float* acc = (float*)d_ws;

    cl_init<<<1, 32, 0, stream>>>(acc);

    // Pass 1: dist1[b,n] = min_m dist, weighted by objectness.
    cl_chamfer<int><<<(BATCH * NPRED) / 128, 256, 0, stream>>>(
        pred, gt, obj, acc + 0, NPRED, MGT);

    // Pass 2 (roles swapped): dist2[b,m] = min_n dist, weighted by mask.
    cl_chamfer<float><<<(BATCH * MGT) / 128, 256, 0, stream>>>(
        gt, pred, mask, acc + 1, MGT, NPRED);

    cl_final<<<1, 256, 0, stream>>>(acc, obj, mask, out);
}